// RenderLayer_26834955665850
// MI455X (gfx1250) — compile-verified
//
#include <hip/hip_runtime.h>

typedef __attribute__((ext_vector_type(2))) float v2f;
typedef __attribute__((ext_vector_type(8))) float v8f;

#define W_IMG 128
#define H_IMG 128
#define NVERT 512
#define NFACE 1024
#define NEAR_P 0.1f
#define FAR_P 10.0f
#define EPS_P 1e-8f
#define NCHUNK 4
#define TRI_PER_CHUNK (NFACE / NCHUNK)
#define NPIX (W_IMG * H_IMG)

// workspace layout in 32-bit words
#define WS_VSX   0
#define WS_VSY   (WS_VSX + NVERT)
#define WS_VZ    (WS_VSY + NVERT)
#define WS_VIW   (WS_VZ + NVERT)
#define WS_VVAL  (WS_VIW + NVERT)
#define WS_TRI   (WS_VVAL + NVERT)          // NFACE * 12 floats
#define WS_CD    (WS_TRI + NFACE * 12)      // NCHUNK * NPIX
#define WS_CB0   (WS_CD + NCHUNK * NPIX)
#define WS_CB1   (WS_CB0 + NCHUNK * NPIX)
#define WS_CB2   (WS_CB1 + NCHUNK * NPIX)
#define WS_CI    (WS_CB2 + NCHUNK * NPIX)   // ints

// -------- Kernel 1: build view@proj, WMMA vertex transform, per-vertex post --------
// One block, 128 threads = 4 wave32s. Each wave does 8 x V_WMMA_F32_16X16X4_F32,
// transforming 16 vertices per WMMA (A = vh tile 16x4, B = M in columns 0..3).
__global__ void vertex_xform_kernel(const float* __restrict__ v,
                                    const float* __restrict__ cam_f,
                                    const float* __restrict__ cam_c,
                                    const float* __restrict__ cam_t,
                                    const float* __restrict__ cam_rt,
                                    float* __restrict__ ws) {
  __shared__ float Ms[16];            // combined (view @ proj), row-major
  __shared__ float clipS[NVERT * 4];  // clip-space coords
  const int tid = threadIdx.x;

  if (tid == 0) {
    // Rodrigues
    float rx = cam_rt[0], ry = cam_rt[1], rz = cam_rt[2];
    float theta = sqrtf(rx * rx + ry * ry + rz * rz + 1e-12f);
    float kx = rx / theta, ky = ry / theta, kz = rz / theta;
    float st = sinf(theta), ct = cosf(theta), oc = 1.0f - ct;
    float K[3][3] = {{0.f, -kz, ky}, {kz, 0.f, -kx}, {-ky, kx, 0.f}};
    float K2[3][3];
    for (int i = 0; i < 3; i++)
      for (int j = 0; j < 3; j++) {
        float s = 0.f;
        for (int k = 0; k < 3; k++) s += K[i][k] * K[k][j];
        K2[i][j] = s;
      }
    float R[3][3];
    for (int i = 0; i < 3; i++)
      for (int j = 0; j < 3; j++)
        R[i][j] = ((i == j) ? 1.0f : 0.0f) + st * K[i][j] + oc * K2[i][j];
    // view = [[R^T, 0],[t, 1]]
    float view[4][4];
    for (int i = 0; i < 4; i++)
      for (int j = 0; j < 4; j++) view[i][j] = 0.f;
    for (int i = 0; i < 3; i++)
      for (int j = 0; j < 3; j++) view[i][j] = R[j][i];
    view[3][0] = cam_t[0]; view[3][1] = cam_t[1]; view[3][2] = cam_t[2];
    view[3][3] = 1.0f;
    // perspective (m), proj = m^T
    float fl = 0.5f * (cam_f[0] + cam_f[1]);
    float ccx = cam_c[0], ccy = cam_c[1];
    const float pco = 0.5f;
    float right  = ((float)W_IMG - (ccx + pco)) * (NEAR_P / fl);
    float left   = -(ccx + pco) * (NEAR_P / fl);
    float top    = (ccy + pco) * (NEAR_P / fl);
    float bottom = -((float)H_IMG - ccy + pco) * (NEAR_P / fl);
    float m[4][4];
    for (int i = 0; i < 4; i++)
      for (int j = 0; j < 4; j++) m[i][j] = 0.f;
    m[0][0] = 2.0f * NEAR_P / (right - left);
    m[0][2] = (right + left) / (right - left);
    m[1][1] = 2.0f * NEAR_P / (top - bottom);
    m[1][2] = (top + bottom) / (top - bottom);
    m[2][2] = -(FAR_P + NEAR_P) / (FAR_P - NEAR_P);
    m[2][3] = -2.0f * FAR_P * NEAR_P / (FAR_P - NEAR_P);
    m[3][2] = -1.0f;
    // final[i][j] = sum_k view[i][k] * proj[k][j],  proj[k][j] = m[j][k]
    for (int i = 0; i < 4; i++)
      for (int j = 0; j < 4; j++) {
        float s = 0.f;
        for (int k = 0; k < 4; k++) s += view[i][k] * m[j][k];
        Ms[i * 4 + j] = s;
      }
  }
  __syncthreads();

  const int lane = tid & 31;
  const int wave = tid >> 5;
  const int col = lane & 15;
  const bool hi = lane >= 16;  // lanes 16-31 hold K=2,3 for A; rows 2,3 for B

  // B operand: 4x16, M in columns 0..3, zero elsewhere (branchless, in-bounds loads)
  const int cc = col & 3;
  float b0v = (col < 4) ? Ms[(hi ? 2 : 0) * 4 + cc] : 0.0f;
  float b1v = (col < 4) ? Ms[(hi ? 3 : 1) * 4 + cc] : 0.0f;
  v2f bmat;
  bmat.x = b0v;
  bmat.y = b1v;

  for (int t8 = 0; t8 < 8; ++t8) {
    int base = wave * 128 + t8 * 16;
    int vi = base + col;
    float vx = v[vi * 3 + 0];
    float vy = v[vi * 3 + 1];
    float vz = v[vi * 3 + 2];
    v2f amat;  // A 16x4 f32: lanes 0-15 -> K0,K1 ; lanes 16-31 -> K2,K3 (vh = x,y,z,1)
    amat.x = hi ? vz : vx;
    amat.y = hi ? 1.0f : vy;
    v8f acc = {};
    v8f d = __builtin_amdgcn_wmma_f32_16x16x4_f32(false, amat, false, bmat,
                                                  (short)0, acc, false, false);
    // D layout: VGPR r, lanes 0-15 -> row base+r, lanes 16-31 -> row base+r+8; col = lane&15
    if (col < 4) {
      int vb = base + (hi ? 8 : 0);
#pragma unroll
      for (int r = 0; r < 8; ++r) clipS[(vb + r) * 4 + col] = d[r];
    }
  }
  __syncthreads();

  for (int i = tid; i < NVERT; i += blockDim.x) {
    float cxx = clipS[i * 4 + 0];
    float cyy = clipS[i * 4 + 1];
    float czz = clipS[i * 4 + 2];
    float cw = clipS[i * 4 + 3];
    bool valid = cw > EPS_P;
    float wsafe = valid ? cw : 1.0f;
    float nx = cxx / wsafe;
    float ny = cyy / wsafe;
    float nz = czz / wsafe;
    ws[WS_VSX + i] = (nx * 0.5f + 0.5f) * (float)W_IMG;
    ws[WS_VSY + i] = (0.5f - ny * 0.5f) * (float)H_IMG;
    ws[WS_VZ + i] = nz;
    ws[WS_VIW + i] = 1.0f / wsafe;
    ws[WS_VVAL + i] = valid ? 1.0f : 0.0f;
  }
}

// -------- Kernel 2: per-triangle setup --------
__global__ void tri_setup_kernel(const int* __restrict__ f, float* __restrict__ ws) {
  int t = blockIdx.x * blockDim.x + threadIdx.x;
  if (t >= NFACE) return;
  int i0 = f[t * 3 + 0], i1 = f[t * 3 + 1], i2 = f[t * 3 + 2];
  float ax = ws[WS_VSX + i0], ay = ws[WS_VSY + i0];
  float bx = ws[WS_VSX + i1], by = ws[WS_VSY + i1];
  float cx = ws[WS_VSX + i2], cy = ws[WS_VSY + i2];
  float area = (bx - ax) * (cy - ay) - (by - ay) * (cx - ax);
  bool areaok = fabsf(area) > EPS_P;
  float asafe = areaok ? area : 1.0f;
  float ia = 1.0f / asafe;
  bool valid = (ws[WS_VVAL + i0] != 0.f) && (ws[WS_VVAL + i1] != 0.f) &&
               (ws[WS_VVAL + i2] != 0.f);
  float flag = (valid && areaok) ? 1.0f : 0.0f;
  float* r = ws + WS_TRI + t * 12;
  r[0] = ax; r[1] = ay; r[2] = bx; r[3] = by; r[4] = cx; r[5] = cy;
  r[6] = ia;
  r[7] = ws[WS_VZ + i0]; r[8] = ws[WS_VZ + i1]; r[9] = ws[WS_VZ + i2];
  r[10] = flag; r[11] = 0.f;
}

// -------- Kernel 3: raster pass 1 (chunked triangle scan, LDS-staged) --------
__global__ void raster_pass1(float* __restrict__ ws) {
  __shared__ float tl[TRI_PER_CHUNK * 12];
  const int tid = threadIdx.x;
  const int chunk = blockIdx.y;
  const float* recs = ws + WS_TRI + chunk * TRI_PER_CHUNK * 12;
  for (int k = tid; k < TRI_PER_CHUNK * 12; k += blockDim.x) tl[k] = recs[k];
  __syncthreads();

  const int p = blockIdx.x * blockDim.x + tid;
  const float px = (float)(p & (W_IMG - 1)) + 0.5f;
  const float py = (float)(p >> 7) + 0.5f;

  float bd = __builtin_inff();
  float bb0 = 0.f, bb1 = 0.f, bb2 = 0.f;
  int bj = 0;
  for (int j = 0; j < TRI_PER_CHUNK; ++j) {
    const float* r = &tl[j * 12];  // uniform address -> LDS broadcast reads
    float ax = r[0], ay = r[1], bx = r[2], by = r[3], cx = r[4], cy = r[5];
    float ia = r[6], za = r[7], zb = r[8], zc = r[9], flag = r[10];
    float w0 = (cx - bx) * (py - by) - (cy - by) * (px - bx);
    float w1 = (ax - cx) * (py - cy) - (ay - cy) * (px - cx);
    float w2 = (bx - ax) * (py - ay) - (by - ay) * (px - ax);
    float b0 = w0 * ia, b1 = w1 * ia, b2 = w2 * ia;
    float z = b0 * za + b1 * zb + b2 * zc;
    bool inside = (b0 >= 0.f) && (b1 >= 0.f) && (b2 >= 0.f) &&
                  (flag != 0.f) && (z >= -1.f) && (z <= 1.f);
    if (inside && (z < bd)) {  // strict < keeps first index (argmin semantics)
      bd = z; bb0 = b0; bb1 = b1; bb2 = b2; bj = j;
    }
  }
  const int ci = chunk * NPIX + p;
  ws[WS_CD + ci] = bd;
  ws[WS_CB0 + ci] = bb0;
  ws[WS_CB1 + ci] = bb1;
  ws[WS_CB2 + ci] = bb2;
  ((int*)ws)[WS_CI + ci] = chunk * TRI_PER_CHUNK + bj;
}

// -------- Kernel 4: reduce chunks in order + perspective-correct shading --------
__global__ void raster_pass2(const int* __restrict__ f, const float* __restrict__ vc,
                             const float* __restrict__ bg, const float* __restrict__ ws,
                             float* __restrict__ out) {
  const int p = blockIdx.x * blockDim.x + threadIdx.x;
  float bd = __builtin_inff();
  float b0 = 0.f, b1 = 0.f, b2 = 0.f;
  int ti = 0;
  for (int c = 0; c < NCHUNK; ++c) {  // chunk order == index order -> first-min wins
    int ci = c * NPIX + p;
    float d = ws[WS_CD + ci];
    if (d < bd) {
      bd = d;
      b0 = ws[WS_CB0 + ci];
      b1 = ws[WS_CB1 + ci];
      b2 = ws[WS_CB2 + ci];
      ti = ((const int*)ws)[WS_CI + ci];
    }
  }
  bool covered = bd <= 1.0f;
  float o0 = bg[0], o1 = bg[1], o2 = bg[2];
  if (covered) {
    int i0 = f[ti * 3 + 0], i1 = f[ti * 3 + 1], i2 = f[ti * 3 + 2];
    float s0 = b0 * ws[WS_VIW + i0];
    float s1 = b1 * ws[WS_VIW + i1];
    float s2 = b2 * ws[WS_VIW + i2];
    float den = s0 + s1 + s2;
    den = (fabsf(den) > EPS_P) ? den : 1.0f;
    o0 = (s0 * vc[i0 * 3 + 0] + s1 * vc[i1 * 3 + 0] + s2 * vc[i2 * 3 + 0]) / den;
    o1 = (s0 * vc[i0 * 3 + 1] + s1 * vc[i1 * 3 + 1] + s2 * vc[i2 * 3 + 1]) / den;
    o2 = (s0 * vc[i0 * 3 + 2] + s1 * vc[i1 * 3 + 2] + s2 * vc[i2 * 3 + 2]) / den;
  }
  out[p * 3 + 0] = o0;
  out[p * 3 + 1] = o1;
  out[p * 3 + 2] = o2;
}

extern "C" void kernel_launch(void* const* d_in, const int* in_sizes, int n_in,
                              void* d_out, int out_size, void* d_ws, size_t ws_size,
                              hipStream_t stream) {
  const float* v      = (const float*)d_in[0];
  const float* vc     = (const float*)d_in[1];
  const int*   f      = (const int*)d_in[2];
  const float* bg     = (const float*)d_in[3];
  const float* cam_f  = (const float*)d_in[4];
  const float* cam_c  = (const float*)d_in[5];
  const float* cam_t  = (const float*)d_in[6];
  const float* cam_rt = (const float*)d_in[7];
  float* ws  = (float*)d_ws;
  float* out = (float*)d_out;

  vertex_xform_kernel<<<1, 128, 0, stream>>>(v, cam_f, cam_c, cam_t, cam_rt, ws);
  tri_setup_kernel<<<(NFACE + 255) / 256, 256, 0, stream>>>(f, ws);
  dim3 g1(NPIX / 256, NCHUNK);
  raster_pass1<<<g1, 256, 0, stream>>>(ws);
  raster_pass2<<<NPIX / 256, 256, 0, stream>>>(f, vc, bg, ws, out);
}